// QModel_65481071408807
// MI455X (gfx1250) — compile-verified
//
#include <hip/hip_runtime.h>

typedef float v2f __attribute__((ext_vector_type(2)));
typedef float v8f __attribute__((ext_vector_type(8)));
typedef int   v4i __attribute__((ext_vector_type(4)));

#define DIMLOG 18
#define NBATCH 64
#define LOWB   12                    // flat bits 0..11 handled in pass A (wires 6..17)
#define CHUNK  (1 << LOWB)           // 4096 complex amps per pass-A block
#define NCHUNK (1 << (DIMLOG - LOWB))// 64 chunks per batch
#define ROWS   (CHUNK / 16)          // 256 LDS rows of 16 cols (low 4 bits = WMMA cols)
#define RSTR   20                    // padded LDS row stride (floats): conflict-free, 16B-aligned

#define BH 64                        // pass-B: all 2^6 high combos (flat bits 12..17, wires 0..5)
#define BC 64                        // contiguous low elements per pass-B block
#define BGROUPS (CHUNK / BC)         // 64 low groups -> 64*64 = 4096 pass-B blocks

// ---- gfx1250 async global<->LDS copies (ASYNCcnt-tracked), if the toolchain
// exposes them; otherwise fall back to sync float4 copies through VGPRs.
#if defined(__has_builtin)
#if __has_builtin(__builtin_amdgcn_global_load_async_to_lds_b128) && \
    __has_builtin(__builtin_amdgcn_global_store_async_from_lds_b128) && \
    __has_builtin(__builtin_amdgcn_s_wait_asynccnt)
#define USE_ASYNC_LDS 1
#endif
#endif

// toolchain declares the async-LDS builtins with generic int4* operands
#define TO_GLOBAL(p) ((v4i*)(p))
#define TO_LDS(p)    ((v4i*)(p))

// suffix-XOR over bits = inverse Gray code (linear over GF(2))
__device__ __forceinline__ int invgray(int x) {
  x ^= x >> 1; x ^= x >> 2; x ^= x >> 4; x ^= x >> 8; x ^= x >> 16;
  return x;
}

// ---------------------------------------------------------------------------
// Setup: per-layer 2x2 complex gates U = RZ*RY, and the fused 16x16 complex
// matrix W = U_w14 (x) U_w15 (x) U_w16 (x) U_w17 for the WMMA stage.
// gates layout: [layer*18 + wire]*8 = {u00r,u00i,u01r,u01i,u10r,u10i,u11r,u11i}
// wmat  layout: per layer 512 floats: Wre[r*16+c] then Wim[r*16+c]
// ---------------------------------------------------------------------------
__global__ __launch_bounds__(256) void qsim_setup(
    const float* __restrict__ params, float* __restrict__ gates,
    float* __restrict__ wmat)
{
  const int tid = threadIdx.x;
  if (tid < 54) {
    const int l = tid / 18, w = tid % 18;
    const float hy = 0.5f * params[l * 36 + w];
    const float hz = 0.5f * params[l * 36 + 18 + w];
    const float cy = __cosf(hy), sy = __sinf(hy);
    const float cz = __cosf(hz), sz = __sinf(hz);
    float* g = gates + (l * 18 + w) * 8;
    g[0] =  cy * cz; g[1] = -cy * sz;   // u00 = cy * e^{-i hz}
    g[2] = -sy * cz; g[3] =  sy * sz;   // u01 = -sy * e^{-i hz}
    g[4] =  sy * cz; g[5] =  sy * sz;   // u10 = sy * e^{+i hz}
    g[6] =  cy * cz; g[7] =  cy * sz;   // u11 = cy * e^{+i hz}
  }
  // fused 16x16 matrix entry (r,c) = tid: Kron over wires 14..17 (bit 3-t of r/c)
  const int r = tid >> 4, c = tid & 15;
#pragma unroll
  for (int l = 0; l < 3; ++l) {
    float ar = 1.0f, ai = 0.0f;
#pragma unroll
    for (int t = 0; t < 4; ++t) {
      const int w = 14 + t;
      const int rb = (r >> (3 - t)) & 1, cb = (c >> (3 - t)) & 1;
      const float hy = 0.5f * params[l * 36 + w];
      const float hz = 0.5f * params[l * 36 + 18 + w];
      const float cy = __cosf(hy), sy = __sinf(hy);
      const float cz = __cosf(hz), sz = __sinf(hz);
      const float amp = (rb == 0) ? (cb == 0 ? cy : -sy) : (cb == 0 ? sy : cy);
      const float er = amp * cz;
      const float ei = (rb == 0) ? (-amp * sz) : (amp * sz);
      const float nr = ar * er - ai * ei, ni = ar * ei + ai * er;
      ar = nr; ai = ni;
    }
    wmat[l * 512 + r * 16 + c]       = ar;
    wmat[l * 512 + 256 + r * 16 + c] = ai;
  }
}

// ---------------------------------------------------------------------------
// Pass A: contiguous 4096-amp chunk in LDS. Applies wires 6..13 as LDS
// butterflies and wires 14..17 as a 16x16 complex GEMM via v_wmma_f32_16x16x4.
// doPerm folds the previous layer's CNOT chain (Gray gather) into addressing:
// chunk G = H^(H>>1) is loaded, intra-chunk inverse-Gray placement into LDS.
// ---------------------------------------------------------------------------
__global__ __launch_bounds__(256) void qsim_pass_low(
    const float* __restrict__ srcRe, const float* __restrict__ srcIm,
    float* __restrict__ dstRe, float* __restrict__ dstIm,
    const float* __restrict__ gates,   // this layer's 18 gates
    const float* __restrict__ wmat,    // this layer's Wre[256],Wim[256]
    int doPerm)
{
  __shared__ float lre[ROWS * RSTR];
  __shared__ float lim[ROWS * RSTR];
  const int tid   = threadIdx.x;
  const int H     = blockIdx.x & (NCHUNK - 1);
  const int batch = blockIdx.x >> 6;
  const int G     = doPerm ? (H ^ (H >> 1)) : H;
  const size_t ibase = ((size_t)batch << DIMLOG) + ((size_t)G << LOWB);
  const size_t obase = ((size_t)batch << DIMLOG) + ((size_t)H << LOWB);
  const int hx = (doPerm && (H & 1)) ? (1 << (LOWB - 1)) : 0;

  // ---- coalesced load; inverse-Gray scatter into LDS when folding the perm
  if (!doPerm) {
#ifdef USE_ASYNC_LDS
#pragma unroll
    for (int r = 0; r < CHUNK / 4 / 256; ++r) {
      const int p = (r * 256 + tid) * 4;
      const int a = (p >> 4) * RSTR + (p & 15);
      __builtin_amdgcn_global_load_async_to_lds_b128(TO_GLOBAL(srcRe + ibase + p), TO_LDS(lre + a), 0, 0);
      __builtin_amdgcn_global_load_async_to_lds_b128(TO_GLOBAL(srcIm + ibase + p), TO_LDS(lim + a), 0, 0);
    }
    __builtin_amdgcn_s_wait_asynccnt(0);
#else
#pragma unroll
    for (int r = 0; r < CHUNK / 4 / 256; ++r) {
      const int p = (r * 256 + tid) * 4;
      const int a = (p >> 4) * RSTR + (p & 15);
      *(float4*)(lre + a) = *(const float4*)(srcRe + ibase + p);
      *(float4*)(lim + a) = *(const float4*)(srcIm + ibase + p);
    }
#endif
  } else {
#pragma unroll
    for (int r = 0; r < CHUNK / 4 / 256; ++r) {
      const int p = (r * 256 + tid) * 4;       // physical offset within chunk
      const float4 vr = *(const float4*)(srcRe + ibase + p);
      const float4 vi = *(const float4*)(srcIm + ibase + p);
      const int L0  = invgray(p ^ hx) & (CHUNK - 1); // virtual index of component 0
      const int grp = L0 & ~3;                       // 4 comps land in one aligned group
      const int a   = (grp >> 4) * RSTR + (grp & 15);
      if ((L0 & 3) == 0) { // d -> {0,1,3,2}
        *(float4*)(lre + a) = make_float4(vr.x, vr.y, vr.w, vr.z);
        *(float4*)(lim + a) = make_float4(vi.x, vi.y, vi.w, vi.z);
      } else {             // L0&3==3: d -> {3,2,0,1}
        *(float4*)(lre + a) = make_float4(vr.z, vr.w, vr.y, vr.x);
        *(float4*)(lim + a) = make_float4(vi.z, vi.w, vi.y, vi.x);
      }
    }
  }
  __syncthreads();

  // ---- butterflies for flat bits 4..11 (wires 13..6), pairs disjoint per stage
  for (int j = 4; j <= 11; ++j) {
    const int w = 17 - j;
    const float* g = gates + w * 8;
    const float u00r = g[0], u00i = g[1], u01r = g[2], u01i = g[3];
    const float u10r = g[4], u10i = g[5], u11r = g[6], u11i = g[7];
#pragma unroll
    for (int r = 0; r < (CHUNK / 2) / 256; ++r) {
      const int p  = r * 256 + tid;
      const int i0 = ((p >> j) << (j + 1)) | (p & ((1 << j) - 1));
      const int i1 = i0 | (1 << j);
      const int a0 = (i0 >> 4) * RSTR + (i0 & 15);
      const int a1 = (i1 >> 4) * RSTR + (i1 & 15);
      const float x0r = lre[a0], x0i = lim[a0];
      const float x1r = lre[a1], x1i = lim[a1];
      lre[a0] = u00r * x0r - u00i * x0i + u01r * x1r - u01i * x1i;
      lim[a0] = u00r * x0i + u00i * x0r + u01r * x1i + u01i * x1r;
      lre[a1] = u10r * x0r - u10i * x0i + u11r * x1r - u11i * x1i;
      lim[a1] = u10r * x0i + u10i * x0r + u11r * x1i + u11i * x1r;
    }
    __syncthreads();
  }

  // ---- wires 14..17 as New(256x16) = Old(256x16) x W^T, complex, via WMMA.
  // NewRe = OldRe*Wr - OldIm*Wi ; NewIm = OldRe*Wi + OldIm*Wr
  // B[k][n] = W[n][k]; chain K=16 as four 16x16x4 steps.
  {
    const int lane = tid & 31;
    const int wave = tid >> 5;          // 8 waves, 16 tiles -> 2 tiles/wave
    const int n    = lane & 15;
    const int kh   = lane >> 4;         // K half select per ISA A/B layout
    v2f brx[4], bix[4], nbx[4];
#pragma unroll
    for (int c4 = 0; c4 < 4; ++c4) {
      const int k0 = c4 * 4 + 2 * kh;
      v2f br; br.x = wmat[n * 16 + k0];       br.y = wmat[n * 16 + k0 + 1];
      v2f bi; bi.x = wmat[256 + n * 16 + k0]; bi.y = wmat[256 + n * 16 + k0 + 1];
      brx[c4] = br; bix[c4] = bi; nbx[c4] = -bi;
    }
    for (int t = wave; t < 16; t += 8) {
      const int rowM = t * 16 + n;      // A fragment: M = lane&15
      v8f accR = {}; v8f accI = {};
#pragma unroll
      for (int c4 = 0; c4 < 4; ++c4) {
        const int k0 = c4 * 4 + 2 * kh;
        const v2f ar = *(const v2f*)(lre + rowM * RSTR + k0);
        const v2f ai = *(const v2f*)(lim + rowM * RSTR + k0);
        accR = __builtin_amdgcn_wmma_f32_16x16x4_f32(false, ar, false, brx[c4], (short)0, accR, false, false);
        accR = __builtin_amdgcn_wmma_f32_16x16x4_f32(false, ai, false, nbx[c4], (short)0, accR, false, false);
        accI = __builtin_amdgcn_wmma_f32_16x16x4_f32(false, ar, false, bix[c4], (short)0, accI, false, false);
        accI = __builtin_amdgcn_wmma_f32_16x16x4_f32(false, ai, false, brx[c4], (short)0, accI, false, false);
      }
      // D layout: VGPR v -> row v + 8*kh, col = lane&15 (all rows owned by this wave)
#pragma unroll
      for (int v = 0; v < 8; ++v) {
        const int row = t * 16 + v + 8 * kh;
        lre[row * RSTR + n] = accR[v];
        lim[row * RSTR + n] = accI[v];
      }
    }
  }
  __syncthreads();

  // ---- coalesced store at the (virtual) output chunk H
#ifdef USE_ASYNC_LDS
#pragma unroll
  for (int r = 0; r < CHUNK / 4 / 256; ++r) {
    const int p = (r * 256 + tid) * 4;
    const int a = (p >> 4) * RSTR + (p & 15);
    __builtin_amdgcn_global_store_async_from_lds_b128(TO_GLOBAL(dstRe + obase + p), TO_LDS(lre + a), 0, 0);
    __builtin_amdgcn_global_store_async_from_lds_b128(TO_GLOBAL(dstIm + obase + p), TO_LDS(lim + a), 0, 0);
  }
  __builtin_amdgcn_s_wait_asynccnt(0);
#else
#pragma unroll
  for (int r = 0; r < CHUNK / 4 / 256; ++r) {
    const int p = (r * 256 + tid) * 4;
    const int a = (p >> 4) * RSTR + (p & 15);
    *(float4*)(dstRe + obase + p) = *(const float4*)(lre + a);
    *(float4*)(dstIm + obase + p) = *(const float4*)(lim + a);
  }
#endif
}

// ---------------------------------------------------------------------------
// Pass B: strided ownership — all 64 high combos (flat bits 12..17, wires 0..5)
// x 64 contiguous low elements. In-place butterflies; optionally fuses the
// final per-wire <Z> + head contraction:  partial = sum |psi|^2 * c(invgray(n))
// ---------------------------------------------------------------------------
__global__ __launch_bounds__(256) void qsim_pass_high(
    float* __restrict__ sRe, float* __restrict__ sIm,
    const float* __restrict__ gates,
    const float* __restrict__ headw, float* __restrict__ partials,
    int finalize)
{
  __shared__ float lre[BH * BC];
  __shared__ float lim[BH * BC];
  const int tid   = threadIdx.x;
  const int grp   = blockIdx.x & (BGROUPS - 1);
  const int batch = blockIdx.x >> 6;
  const size_t base = ((size_t)batch << DIMLOG);
  const int clow = grp * BC;

#ifdef USE_ASYNC_LDS
#pragma unroll
  for (int r = 0; r < (BH * BC) / 4 / 256; ++r) {
    const int e = (r * 256 + tid) * 4;
    const int h = e >> 6, c = e & 63;
    const size_t gi = base + ((size_t)h << LOWB) + clow + c;
    __builtin_amdgcn_global_load_async_to_lds_b128(TO_GLOBAL(sRe + gi), TO_LDS(lre + e), 0, 0);
    __builtin_amdgcn_global_load_async_to_lds_b128(TO_GLOBAL(sIm + gi), TO_LDS(lim + e), 0, 0);
  }
  __builtin_amdgcn_s_wait_asynccnt(0);
#else
#pragma unroll
  for (int r = 0; r < (BH * BC) / 4 / 256; ++r) {
    const int e = (r * 256 + tid) * 4;
    const int h = e >> 6, c = e & 63;
    const size_t gi = base + ((size_t)h << LOWB) + clow + c;
    *(float4*)(lre + e) = *(const float4*)(sRe + gi);
    *(float4*)(lim + e) = *(const float4*)(sIm + gi);
  }
#endif
  __syncthreads();

  for (int jh = 0; jh < 6; ++jh) {       // global bit 12+jh  <->  wire 5-jh
    const int w = 5 - jh;
    const int jb = jh + 6;               // bit position within block index
    const float* g = gates + w * 8;
    const float u00r = g[0], u00i = g[1], u01r = g[2], u01i = g[3];
    const float u10r = g[4], u10i = g[5], u11r = g[6], u11i = g[7];
#pragma unroll
    for (int r = 0; r < (BH * BC / 2) / 256; ++r) {
      const int p  = r * 256 + tid;
      const int i0 = ((p >> jb) << (jb + 1)) | (p & ((1 << jb) - 1));
      const int i1 = i0 | (1 << jb);
      const float x0r = lre[i0], x0i = lim[i0];
      const float x1r = lre[i1], x1i = lim[i1];
      lre[i0] = u00r * x0r - u00i * x0i + u01r * x1r - u01i * x1i;
      lim[i0] = u00r * x0i + u00i * x0r + u01r * x1i + u01i * x1r;
      lre[i1] = u10r * x0r - u10i * x0i + u11r * x1r - u11i * x1i;
      lim[i1] = u10r * x0i + u10i * x0r + u11r * x1i + u11i * x1r;
    }
    __syncthreads();
  }

  if (!finalize) {
#ifdef USE_ASYNC_LDS
#pragma unroll
    for (int r = 0; r < (BH * BC) / 4 / 256; ++r) {
      const int e = (r * 256 + tid) * 4;
      const int h = e >> 6, c = e & 63;
      const size_t gi = base + ((size_t)h << LOWB) + clow + c;
      __builtin_amdgcn_global_store_async_from_lds_b128(TO_GLOBAL(sRe + gi), TO_LDS(lre + e), 0, 0);
      __builtin_amdgcn_global_store_async_from_lds_b128(TO_GLOBAL(sIm + gi), TO_LDS(lim + e), 0, 0);
    }
    __builtin_amdgcn_s_wait_asynccnt(0);
#else
#pragma unroll
    for (int r = 0; r < (BH * BC) / 4 / 256; ++r) {
      const int e = (r * 256 + tid) * 4;
      const int h = e >> 6, c = e & 63;
      const size_t gi = base + ((size_t)h << LOWB) + clow + c;
      *(float4*)(sRe + gi) = *(const float4*)(lre + e);
      *(float4*)(sIm + gi) = *(const float4*)(lim + e);
    }
#endif
  } else {
    float hw[18];
#pragma unroll
    for (int i = 0; i < 18; ++i) hw[i] = headw[i];
    float acc = 0.0f;
#pragma unroll
    for (int r = 0; r < (BH * BC) / 256; ++r) {
      const int e = r * 256 + tid;
      const int h = e >> 6, c = e & 63;
      const int o = (h << LOWB) | clow | c;   // physical (pre-final-perm) index
      const int m = invgray(o);               // virtual index after CNOT chain
      const float pr = lre[e], pi = lim[e];
      const float p2 = pr * pr + pi * pi;
      float wgt = 0.0f;
#pragma unroll
      for (int j = 0; j < 18; ++j)            // wire w = 17-j, sign = 1-2*bit
        wgt += ((m >> j) & 1) ? -hw[17 - j] : hw[17 - j];
      acc += p2 * wgt;
    }
    __syncthreads();
    lre[tid] = acc;
    __syncthreads();
    for (int s = 128; s > 0; s >>= 1) {
      if (tid < s) lre[tid] += lre[tid + s];
      __syncthreads();
    }
    if (tid == 0) partials[blockIdx.x] = lre[0];
  }
}

__global__ void qsim_finish(const float* __restrict__ partials,
                            const float* __restrict__ headb,
                            float* __restrict__ out)
{
  const int b = threadIdx.x;  // 64 batches
  float s = headb[0];
  for (int g = 0; g < BGROUPS; ++g) s += partials[b * BGROUPS + g];
  out[b] = s;
}

// ---------------------------------------------------------------------------
extern "C" void kernel_launch(void* const* d_in, const int* in_sizes, int n_in,
                              void* d_out, int out_size, void* d_ws, size_t ws_size,
                              hipStream_t stream) {
  const float* stateRe = (const float*)d_in[0];
  const float* stateIm = (const float*)d_in[1];
  const float* params  = (const float*)d_in[2];
  const float* headw   = (const float*)d_in[3];
  const float* headb   = (const float*)d_in[4];
  float* out = (float*)d_out;

  const size_t N = (size_t)NBATCH << DIMLOG;   // 16,777,216 amps
  float* ws = (float*)d_ws;
  float* Are = ws;             float* Aim = Are + N;
  float* Bre = Aim + N;        float* Bim = Bre + N;
  float* gates    = Bim + N;            // 3*18*8 = 432 floats
  float* wmat     = gates + 432;        // 3*512 = 1536 floats
  float* partials = wmat + 1536;        // 4096 floats

  const dim3 blk(256);
  const dim3 gridA(NBATCH * NCHUNK);    // 4096
  const dim3 gridB(NBATCH * BGROUPS);   // 4096

  qsim_setup<<<1, blk, 0, stream>>>(params, gates, wmat);
  // layer 0
  qsim_pass_low <<<gridA, blk, 0, stream>>>(stateRe, stateIm, Are, Aim, gates + 0,   wmat + 0,    0);
  qsim_pass_high<<<gridB, blk, 0, stream>>>(Are, Aim, gates + 0,   nullptr, nullptr, 0);
  // layer 1 (Gray-gather folds layer-0 CNOT chain)
  qsim_pass_low <<<gridA, blk, 0, stream>>>(Are, Aim, Bre, Bim, gates + 144, wmat + 512,  1);
  qsim_pass_high<<<gridB, blk, 0, stream>>>(Bre, Bim, gates + 144, nullptr, nullptr, 0);
  // layer 2
  qsim_pass_low <<<gridA, blk, 0, stream>>>(Bre, Bim, Are, Aim, gates + 288, wmat + 1024, 1);
  qsim_pass_high<<<gridB, blk, 0, stream>>>(Are, Aim, gates + 288, headw, partials, 1);
  // deterministic final reduction + bias
  qsim_finish<<<1, 64, 0, stream>>>(partials, headb, out);
}